// LossCR_59055800320006
// MI455X (gfx1250) — compile-verified
//
#include <hip/hip_runtime.h>
#include <hip/hip_bf16.h>
#include <math.h>

// ---------------- problem constants (from reference) ----------------
#define NB     4
#define HW     262144        // 512*512 (power of two)
#define NPIX   1048576       // NB*HW
#define DFEAT  128
#define NCLS   21
#define LS     0.1f          // label smoothing
#define LREG   0.4f          // lambda_reg

// workspace float offsets
#define WS_SUMS   0          // 21*128 = 2688
#define WS_CNT    2688       // 21
#define WS_SEM    2720       // 1
#define WS_VALID  2721       // 1
#define WS_TOTAL  4096

typedef float v2f __attribute__((ext_vector_type(2)));
typedef float v4f __attribute__((ext_vector_type(4)));
typedef float v8f __attribute__((ext_vector_type(8)));
typedef int   v4i __attribute__((ext_vector_type(4)));

// Single-instruction LDS float add (no return) — guarantees ds_add_f32, never a
// CAS loop. Flat LDS addresses carry the LDS byte offset in addr[31:0]
// (ISA §10.2 aperture mapping), which is exactly what the VDS ADDR VGPR wants.
__device__ __forceinline__ void ldsAddF32(float* p, float v) {
  unsigned off = (unsigned)(unsigned long long)p;
  asm volatile("ds_add_f32 %0, %1" :: "v"(off), "v"(v) : "memory");
}

// ---------------- init: zero the scratch accumulators ----------------
__global__ void init_ws(float* __restrict__ ws) {
  int i = blockIdx.x * 256 + threadIdx.x;
  if (i < WS_TOTAL) ws[i] = 0.0f;
}

// ---------------- class counts ----------------
// 32 blocks x 256 threads x 32 iters x 4 px (int4 loads)
__global__ void count_kernel(const int* __restrict__ labels,
                             float* __restrict__ ws_counts) {
  __shared__ float cnt[NCLS];
  if (threadIdx.x < NCLS) cnt[threadIdx.x] = 0.0f;
  __syncthreads();
  long long gbase = (long long)blockIdx.x * 8192 + threadIdx.x;
  for (int i = 0; i < 32; ++i) {
    long long q = (gbase + (long long)i * 256) << 2;
    v4i lab = __builtin_nontemporal_load((const v4i*)(labels + q));
#pragma unroll
    for (int j = 0; j < 4; ++j)
      if (lab[j] >= 0 && lab[j] < NCLS) ldsAddF32(&cnt[lab[j]], 1.0f);
  }
  __syncthreads();
  if (threadIdx.x < NCLS) atomicAdd(&ws_counts[threadIdx.x], cnt[threadIdx.x]);
}

// ---------------- segment sum of z by label ----------------
// grid = (128 channels, 64 pixel chunks), block = 256 (8 wave32), 16 iters x 4 px.
// Contention-free LDS histogram: each (wave,lane) owns its own 21-slot column;
// ds_add_f32 with distinct banks per lane -> zero conflicts.
// z streamed once with NT hint (B128 loads); labels re-read per channel stay in L2.
__global__ void seg_sum_kernel(const float* __restrict__ z,
                               const int*   __restrict__ labels,
                               float*       __restrict__ ws_sums) {
  __shared__ float acc[8][NCLS][32];   // 21504 B
  __shared__ float acc2[8][NCLS];
  const int lane = threadIdx.x & 31;
  const int wv   = threadIdx.x >> 5;
#pragma unroll
  for (int c = 0; c < NCLS; ++c) acc[wv][c][lane] = 0.0f;
  __syncthreads();

  const int ch = blockIdx.x;
  const long long gbase = (long long)blockIdx.y * 4096 + threadIdx.x;
  for (int i = 0; i < 16; ++i) {
    long long q = (gbase + (long long)i * 256) << 2;   // 4-aligned pixel index
    int n = (int)(q >> 18);            // q / HW   (groups never straddle n)
    int p = (int)(q & (HW - 1));       // q % HW
    v4i lab = __builtin_nontemporal_load((const v4i*)(labels + q));
    v4f zv  = __builtin_nontemporal_load(
                 (const v4f*)(z + ((long long)(n * DFEAT + ch)) * HW + p));
#pragma unroll
    for (int j = 0; j < 4; ++j)
      if (lab[j] >= 0 && lab[j] < NCLS) ldsAddF32(&acc[wv][lab[j]][lane], zv[j]);
  }
  __syncthreads();

  int t = threadIdx.x;
  if (t < 8 * NCLS) {
    int w = t / NCLS, c = t % NCLS;
    float s = 0.0f;
#pragma unroll
    for (int l = 0; l < 32; ++l) s += acc[w][c][l];
    acc2[w][c] = s;
  }
  __syncthreads();
  if (t < NCLS) {
    float s = 0.0f;
#pragma unroll
    for (int w = 0; w < 8; ++w) s += acc2[w][t];
    atomicAdd(&ws_sums[t * DFEAT + ch], s);   // 21 global atomics / block
  }
}

// ---------------- per-pixel label-smoothed CE over preds ----------------
// 1024 blocks x 256 threads, 4 px per thread via B128 loads (21 planes).
__global__ void sem_kernel(const float* __restrict__ preds,
                           const int*   __restrict__ labels,
                           float* __restrict__ ws_sem,
                           float* __restrict__ ws_valid) {
  long long q = ((long long)blockIdx.x * 256 + threadIdx.x) << 2;
  int n = (int)(q >> 18);
  int p = (int)(q & (HW - 1));
  const float* pp = preds + (long long)n * NCLS * HW + p;
  v4i lab = __builtin_nontemporal_load((const v4i*)(labels + q));
  v4f lg[NCLS];
#pragma unroll
  for (int c = 0; c < NCLS; ++c)
    lg[c] = __builtin_nontemporal_load((const v4f*)(pp + (long long)c * HW));

  float lsum = 0.0f, lcnt = 0.0f;
#pragma unroll
  for (int j = 0; j < 4; ++j) {
    float mx = -INFINITY;
#pragma unroll
    for (int c = 0; c < NCLS; ++c) mx = fmaxf(mx, lg[c][j]);
    float ssum = 0.0f, tot = 0.0f, tl = 0.0f;
#pragma unroll
    for (int c = 0; c < NCLS; ++c) {
      ssum += expf(lg[c][j] - mx);
      tot  += lg[c][j];
      if (c == lab[j]) tl = lg[c][j];
    }
    float lse = mx + logf(ssum);
    if (lab[j] >= 0) {
      // (1-ls)*(lse - logit_t) + ls*(lse - mean) = lse - 0.9*t - 0.1*mean
      lsum += lse - (1.0f - LS) * tl - LS * (tot * (1.0f / (float)NCLS));
      lcnt += 1.0f;
    }
  }
  __shared__ float red[256];
  red[threadIdx.x] = lsum; __syncthreads();
  for (int s = 128; s > 0; s >>= 1) {
    if (threadIdx.x < s) red[threadIdx.x] += red[threadIdx.x + s];
    __syncthreads();
  }
  if (threadIdx.x == 0) atomicAdd(ws_sem, red[0]);
  __syncthreads();
  red[threadIdx.x] = lcnt; __syncthreads();
  for (int s = 128; s > 0; s >>= 1) {
    if (threadIdx.x < s) red[threadIdx.x] += red[threadIdx.x + s];
    __syncthreads();
  }
  if (threadIdx.x == 0) atomicAdd(ws_valid, red[0]);
}

// ---------------- finalize: Z_bar @ W_star via WMMA f32 16x16x4 + CE ----------------
// Single wave32. 21x21 logits = 2x2 tiles of 16x16, K=128 in 32 steps of K=4.
__global__ void finalize_kernel(const float* __restrict__ ws_sums,
                                const float* __restrict__ ws_counts,
                                const float* __restrict__ W_star,
                                const float* __restrict__ ws_sem,
                                const float* __restrict__ ws_valid,
                                float* __restrict__ out) {
  __shared__ float Zb[32][DFEAT];    // rows >= 21 zero-padded
  __shared__ float Wl[DFEAT][32];    // cols >= 21 zero-padded
  __shared__ float logits[32][32];
  __shared__ float rowloss[NCLS];
  const int lane = threadIdx.x;      // 0..31

  for (int idx = lane; idx < 32 * DFEAT; idx += 32) {
    int r = idx >> 7, k = idx & 127;
    float v = 0.0f;
    if (r < NCLS) {
      float cnt = ws_counts[r];
      if (cnt > 0.0f) v = ws_sums[r * DFEAT + k] / fmaxf(cnt, 1.0f);
    }
    Zb[r][k] = v;
  }
  for (int idx = lane; idx < DFEAT * 32; idx += 32) {
    int k = idx >> 5, nc = idx & 31;
    Wl[k][nc] = (nc < NCLS) ? W_star[k * NCLS + nc] : 0.0f;
  }
  __syncthreads();

  // ISA 7.12.2 f32 16x16x4 layout: lanes 0-15 carry K=k0,k0+1 ; lanes 16-31 K=k0+2,k0+3
  const int hs = lane >> 4;          // half select
  const int lr = lane & 15;
  v8f acc00 = {}, acc01 = {}, acc10 = {}, acc11 = {};
  for (int k0 = 0; k0 < DFEAT; k0 += 4) {
    int ka = k0 + hs * 2;
    v2f a0, a1, b0, b1;
    a0.x = Zb[lr][ka];       a0.y = Zb[lr][ka + 1];        // M tile 0
    a1.x = Zb[16 + lr][ka];  a1.y = Zb[16 + lr][ka + 1];   // M tile 1
    b0.x = Wl[ka][lr];       b0.y = Wl[ka + 1][lr];        // N tile 0
    b1.x = Wl[ka][16 + lr];  b1.y = Wl[ka + 1][16 + lr];   // N tile 1
    acc00 = __builtin_amdgcn_wmma_f32_16x16x4_f32(false, a0, false, b0, (short)0, acc00, false, false);
    acc01 = __builtin_amdgcn_wmma_f32_16x16x4_f32(false, a0, false, b1, (short)0, acc01, false, false);
    acc10 = __builtin_amdgcn_wmma_f32_16x16x4_f32(false, a1, false, b0, (short)0, acc10, false, false);
    acc11 = __builtin_amdgcn_wmma_f32_16x16x4_f32(false, a1, false, b1, (short)0, acc11, false, false);
  }
  // C/D layout: element v of lane L -> M = v + 8*(L>=16), N = L%16 (per tile)
#pragma unroll
  for (int v = 0; v < 8; ++v) {
    logits[v + 8 * hs][lr]           = acc00[v];
    logits[v + 8 * hs][16 + lr]      = acc01[v];
    logits[16 + v + 8 * hs][lr]      = acc10[v];
    logits[16 + v + 8 * hs][16 + lr] = acc11[v];
  }
  __syncthreads();

  if (lane < NCLS) {
    float mx = -INFINITY, ssum = 0.0f, tot = 0.0f;
    for (int c = 0; c < NCLS; ++c) mx = fmaxf(mx, logits[lane][c]);
    for (int c = 0; c < NCLS; ++c) { ssum += expf(logits[lane][c] - mx); tot += logits[lane][c]; }
    float lse = mx + logf(ssum);
    rowloss[lane] = lse - (1.0f - LS) * logits[lane][lane] - LS * (tot / (float)NCLS);
  }
  __syncthreads();
  if (lane == 0) {
    float lcr = 0.0f;
    for (int r = 0; r < NCLS; ++r) lcr += rowloss[r];
    lcr /= (float)NCLS;
    out[0] = LREG * lcr + ws_sem[0] / fmaxf(ws_valid[0], 1.0f);
  }
}

// ---------------- launch ----------------
extern "C" void kernel_launch(void* const* d_in, const int* in_sizes, int n_in,
                              void* d_out, int out_size, void* d_ws, size_t ws_size,
                              hipStream_t stream) {
  const float* preds  = (const float*)d_in[0];
  const int*   labels = (const int*)  d_in[1];
  // d_in[2] = labels_depth (unused by reference)
  const float* z      = (const float*)d_in[3];
  const float* W_star = (const float*)d_in[4];
  float* ws     = (float*)d_ws;
  float* sums   = ws + WS_SUMS;
  float* counts = ws + WS_CNT;
  float* sem    = ws + WS_SEM;
  float* valid  = ws + WS_VALID;

  init_ws<<<16, 256, 0, stream>>>(ws);
  count_kernel<<<32, 256, 0, stream>>>(labels, counts);
  seg_sum_kernel<<<dim3(DFEAT, 64), 256, 0, stream>>>(z, labels, sums);
  sem_kernel<<<1024, 256, 0, stream>>>(preds, labels, sem, valid);
  finalize_kernel<<<1, 32, 0, stream>>>(sums, counts, W_star, sem, valid, (float*)d_out);
}